// SS2D_18253611008836
// MI455X (gfx1250) — compile-verified
//
#include <hip/hip_runtime.h>
#include <math.h>

// ---------------- model constants ----------------
#define D_MODEL   256
#define D_STATE   16
#define L_SEQ     12
#define D_INNER   512
#define DT_RANK   16
#define B_SEQ     2
#define HH        32
#define WW        32
#define BT        (B_SEQ * L_SEQ)        // 24
#define M_ROWS    (BT * HH * WW)         // 24576
#define P_ROWS    (B_SEQ * HH * WW)      // 2048
#define C_DBL     (DT_RANK + 2 * D_STATE) // 48
#define LN_EPS    1e-5f

typedef __attribute__((ext_vector_type(2))) float v2f;
typedef __attribute__((ext_vector_type(4))) float v4f;
typedef __attribute__((ext_vector_type(8))) float v8f;

__device__ __forceinline__ float fast_silu(float x) {
    return x / (1.0f + __expf(-x));
}
__device__ __forceinline__ float fast_softplus(float x) {
    if (x > 20.0f) return x;
    return __logf(1.0f + __expf(x));
}

// =====================================================================
// Tiled fp32-WMMA GEMM body:  C[M,N] = A[M,K] * B[N,K]^T
// Block tile 64x128, BK=16, 256 threads (8 waves, wave tile 16x64).
// LDS double buffered, rows padded to 20 words:
//   - b128 global loads coalesced along contiguous K
//   - 16B-aligned LDS b128 stores, conflict-free ds_load_b64 frag reads
// SPLIT=true: split columns [0,512)/[512,1024) into o0/o1 (in_proj).
// =====================================================================
#define BKPAD 20

template<int KLEN, bool SPLIT>
__device__ __forceinline__ void gemm_tiled_body(
    const float* __restrict__ A, const float* __restrict__ Bw,
    float* __restrict__ o0, float* __restrict__ o1, const int ncols)
{
    __shared__ float shA[2][64][BKPAD];
    __shared__ float shB[2][128][BKPAD];

    const int t      = threadIdx.x;
    const int lane   = t & 31;
    const int wid    = t >> 5;
    const int laneLo = lane & 15;
    const int laneHi = lane >> 4;
    const int waveM  = wid & 3;    // 4 waves along M
    const int waveN  = wid >> 2;   // 2 waves along N

    const int blockM = blockIdx.x * 64;
    const int blockN = blockIdx.y * 128;

    // global staging coords (coalesced along K)
    const int ar = t >> 2, ac = (t & 3) * 4;   // A: 64 rows x 16 cols, 1 b128/thr
    const int br = t >> 1, bc = (t & 1) * 8;   // B: 128 rows x 16 cols, 2 b128/thr
    const float* gA = A  + (size_t)(blockM + ar) * KLEN + ac;
    const float* gB = Bw + (size_t)(blockN + br) * KLEN + bc;

    // prologue: chunk 0 -> buf 0
    v4f ra  = *(const v4f*)(gA);
    v4f rb0 = *(const v4f*)(gB);
    v4f rb1 = *(const v4f*)(gB + 4);
    *(v4f*)&shA[0][ar][ac]     = ra;
    *(v4f*)&shB[0][br][bc]     = rb0;
    *(v4f*)&shB[0][br][bc + 4] = rb1;
    __syncthreads();

    v8f acc[4] = {};
    const int aRow = waveM * 16 + laneLo;
    const int bRow = waveN * 64 + laneLo;

    constexpr int NCHUNK = KLEN / 16;
    #pragma unroll 2
    for (int ch = 0; ch < NCHUNK; ++ch) {
        const int  buf  = ch & 1;
        const bool more = (ch + 1) < NCHUNK;
        if (more) {
            const int kc = (ch + 1) * 16;
            ra  = *(const v4f*)(gA + kc);
            rb0 = *(const v4f*)(gB + kc);
            rb1 = *(const v4f*)(gB + kc + 4);
            if (ch + 2 < NCHUNK) {
                __builtin_prefetch(gA + (ch + 2) * 16, 0, 1);  // global_prefetch_b8
                __builtin_prefetch(gB + (ch + 2) * 16, 0, 1);
            }
        }
        #pragma unroll
        for (int kk2 = 0; kk2 < 16; kk2 += 4) {
            v2f a = *(const v2f*)&shA[buf][aRow][kk2 + 2 * laneHi];
            #pragma unroll
            for (int j = 0; j < 4; ++j) {
                v2f b = *(const v2f*)&shB[buf][bRow + j * 16][kk2 + 2 * laneHi];
                acc[j] = __builtin_amdgcn_wmma_f32_16x16x4_f32(
                    false, a, false, b, (short)0, acc[j], false, false);
            }
        }
        if (more) {
            const int nb = buf ^ 1;
            *(v4f*)&shA[nb][ar][ac]     = ra;
            *(v4f*)&shB[nb][br][bc]     = rb0;
            *(v4f*)&shB[nb][br][bc + 4] = rb1;
        }
        __syncthreads();
    }

    #pragma unroll
    for (int j = 0; j < 4; ++j) {
        const int c = blockN + waveN * 64 + j * 16 + laneLo;
        #pragma unroll
        for (int i = 0; i < 8; ++i) {
            const int m = blockM + waveM * 16 + i + 8 * laneHi;
            const float v = acc[j][i];
            if (SPLIT) {
                if (c < D_INNER) o0[(size_t)m * D_INNER + c] = v;
                else             o1[(size_t)m * D_INNER + (c - D_INNER)] = v;
            } else {
                o0[(size_t)m * ncols + c] = v;
            }
        }
    }
}

// Kernel 1: in_proj  (M=24576, N=1024, K=256), splits xc / z
__global__ __launch_bounds__(256) void k_inproj_gemm(
    const float* __restrict__ X, const float* __restrict__ Win,
    float* __restrict__ xc, float* __restrict__ zg)
{
    gemm_tiled_body<D_MODEL, true>(X, Win, xc, zg, 0);
}

// Kernel 5: out_proj (M=24576, N=256, K=512)
__global__ __launch_bounds__(256) void k_outproj_gemm(
    const float* __restrict__ Y, const float* __restrict__ Wout,
    float* __restrict__ out)
{
    gemm_tiled_body<D_INNER, false>(Y, Wout, out, nullptr, D_MODEL);
}

// =====================================================================
// Kernel 2: depthwise 3x3 conv (SAME, cross-correlation) + bias + SiLU.
// Reads xc[(bt*1024+hw), d], writes scan layout xcs[(p*L + l), d].
// =====================================================================
__global__ __launch_bounds__(512) void k_conv_dw(
    const float* __restrict__ xc,      // [M,512]
    const float* __restrict__ cw,      // [512,1,3,3]
    const float* __restrict__ cb,      // [512]
    float* __restrict__ xcs)           // [P*L,512]
{
    const int d  = threadIdx.x;
    const int bt = blockIdx.x >> 10;
    const int hw = blockIdx.x & 1023;
    const int h  = hw >> 5;
    const int w  = hw & 31;

    float acc = 0.0f;
    #pragma unroll
    for (int kh = 0; kh < 3; ++kh) {
        const int hh = h + kh - 1;
        if (hh < 0 || hh >= HH) continue;
        #pragma unroll
        for (int kw = 0; kw < 3; ++kw) {
            const int ww = w + kw - 1;
            if (ww < 0 || ww >= WW) continue;
            acc += cw[d * 9 + kh * 3 + kw] *
                   xc[((size_t)bt * 1024 + hh * WW + ww) * D_INNER + d];
        }
    }
    acc = fast_silu(acc + cb[d]);

    const int bb = bt / L_SEQ;
    const int l  = bt % L_SEQ;
    const int p  = bb * 1024 + hw;
    xcs[((size_t)p * L_SEQ + l) * D_INNER + d] = acc;
}

// =====================================================================
// Kernel 3: x_dbl GEMM for both directions (N=48: 3 col tiles, K=512).
// Block = 256 thr (8 waves): wave w -> (rowTile = w>>1, k = w&1).
// Register double-buffered fragment feed.
// =====================================================================
__global__ __launch_bounds__(256) void k_xdbl_gemm(
    const float* __restrict__ xcs,     // [M,512]
    const float* __restrict__ Wx,      // [2,48,512]
    float* __restrict__ xdbl0,         // [M,48]
    float* __restrict__ xdbl1)         // [M,48]
{
    const int lane   = threadIdx.x & 31;
    const int wid    = threadIdx.x >> 5;
    const int k      = wid & 1;
    const int laneLo = lane & 15;
    const int laneHi = lane >> 4;
    const int rowBase = blockIdx.x * 64 + (wid >> 1) * 16;

    const float* Wk   = Wx + (size_t)k * C_DBL * D_INNER;
    float*       outk = k ? xdbl1 : xdbl0;
    const float* arow = xcs + (size_t)(rowBase + laneLo) * D_INNER + 2 * laneHi;
    const float* brow = Wk + (size_t)laneLo * D_INNER + 2 * laneHi;

    v8f acc[3] = {};
    v2f a_c = *(const v2f*)(arow);
    v2f b_c[3];
    #pragma unroll
    for (int j = 0; j < 3; ++j)
        b_c[j] = *(const v2f*)(brow + (size_t)j * 16 * D_INNER);

    #pragma unroll 4
    for (int kk = 0; kk < D_INNER - 4; kk += 4) {
        v2f a_n = *(const v2f*)(arow + kk + 4);
        v2f b_n[3];
        #pragma unroll
        for (int j = 0; j < 3; ++j)
            b_n[j] = *(const v2f*)(brow + (size_t)j * 16 * D_INNER + kk + 4);
        #pragma unroll
        for (int j = 0; j < 3; ++j)
            acc[j] = __builtin_amdgcn_wmma_f32_16x16x4_f32(
                false, a_c, false, b_c[j], (short)0, acc[j], false, false);
        a_c = a_n;
        #pragma unroll
        for (int j = 0; j < 3; ++j) b_c[j] = b_n[j];
    }
    #pragma unroll
    for (int j = 0; j < 3; ++j)
        acc[j] = __builtin_amdgcn_wmma_f32_16x16x4_f32(
            false, a_c, false, b_c[j], (short)0, acc[j], false, false);

    #pragma unroll
    for (int j = 0; j < 3; ++j) {
        const int c = j * 16 + laneLo;
        #pragma unroll
        for (int i = 0; i < 8; ++i) {
            const int m = rowBase + i + 8 * laneHi;
            outk[(size_t)m * C_DBL + c] = acc[j][i];
        }
    }
}

// =====================================================================
// Kernel 4: fused dt_proj + softplus + bidirectional selective scan
//           + D*u merge + LayerNorm(d_inner) + SiLU(z) gate.
// Block = one pixel p (2048 blocks), 512 threads = d channel.
// =====================================================================
__global__ __launch_bounds__(512) void k_scan_ln_gate(
    const float* __restrict__ xcs,     // [P*L,512]  u
    const float* __restrict__ xdbl0,   // [P*L,48]
    const float* __restrict__ xdbl1,   // [P*L,48]
    const float* __restrict__ dtw,     // [2,512,16]
    const float* __restrict__ dtb,     // [2,512]
    const float* __restrict__ Alogs,   // [2*512,16]
    const float* __restrict__ Ds,      // [2*512]
    const float* __restrict__ lnw,     // [512]
    const float* __restrict__ lnb,     // [512]
    const float* __restrict__ zg,      // [M,512]
    float* __restrict__ yg)            // [M,512]
{
    __shared__ float sh[2][L_SEQ][C_DBL];
    __shared__ float redx[16], redy[16];
    __shared__ float statx, staty;

    const int p = blockIdx.x;
    const int d = threadIdx.x;

    for (int idx = threadIdx.x; idx < 2 * L_SEQ * C_DBL; idx += 512) {
        const int dir = idx / (L_SEQ * C_DBL);
        const int rem = idx % (L_SEQ * C_DBL);
        const int l   = rem / C_DBL;
        const int c   = rem % C_DBL;
        const float* src = dir ? xdbl1 : xdbl0;
        sh[dir][l][c] = src[((size_t)p * L_SEQ + l) * C_DBL + c];
    }
    __syncthreads();

    float u[L_SEQ];
    #pragma unroll
    for (int l = 0; l < L_SEQ; ++l)
        u[l] = xcs[((size_t)p * L_SEQ + l) * D_INNER + d];

    float ym[L_SEQ];

    // ---- direction 0 (forward) ----
    {
        float A[D_STATE], w16[DT_RANK];
        #pragma unroll
        for (int n = 0; n < D_STATE; ++n) A[n] = -__expf(Alogs[(size_t)d * D_STATE + n]);
        #pragma unroll
        for (int r = 0; r < DT_RANK; ++r) w16[r] = dtw[(size_t)d * DT_RANK + r];
        const float bias = dtb[d];

        float h[D_STATE] = {};
        for (int l = 0; l < L_SEQ; ++l) {
            float dt = bias;
            #pragma unroll
            for (int r = 0; r < DT_RANK; ++r) dt += sh[0][l][r] * w16[r];
            dt = fast_softplus(dt);
            const float du = dt * u[l];
            float yl = 0.0f;
            #pragma unroll
            for (int n = 0; n < D_STATE; ++n) {
                h[n] = __expf(dt * A[n]) * h[n] + du * sh[0][l][DT_RANK + n];
                yl  += h[n] * sh[0][l][DT_RANK + D_STATE + n];
            }
            ym[l] = yl;
        }
    }
    // ---- direction 1 (flipped): walk original l downwards ----
    {
        float A[D_STATE], w16[DT_RANK];
        #pragma unroll
        for (int n = 0; n < D_STATE; ++n) A[n] = -__expf(Alogs[(size_t)(D_INNER + d) * D_STATE + n]);
        #pragma unroll
        for (int r = 0; r < DT_RANK; ++r) w16[r] = dtw[(size_t)(D_INNER + d) * DT_RANK + r];
        const float bias = dtb[D_INNER + d];

        float h[D_STATE] = {};
        for (int l = L_SEQ - 1; l >= 0; --l) {
            float dt = bias;
            #pragma unroll
            for (int r = 0; r < DT_RANK; ++r) dt += sh[1][l][r] * w16[r];
            dt = fast_softplus(dt);
            const float du = dt * u[l];
            float yl = 0.0f;
            #pragma unroll
            for (int n = 0; n < D_STATE; ++n) {
                h[n] = __expf(dt * A[n]) * h[n] + du * sh[1][l][DT_RANK + n];
                yl  += h[n] * sh[1][l][DT_RANK + D_STATE + n];
            }
            ym[l] += yl;
        }
    }

    const float Dsum = Ds[d] + Ds[D_INNER + d];
    #pragma unroll
    for (int l = 0; l < L_SEQ; ++l) ym[l] += Dsum * u[l];

    // ---- LayerNorm over d + SiLU(z) gate, per l ----
    const int bb = p >> 10;
    const int hw = p & 1023;
    const float gw = lnw[d], gb = lnb[d];
    const int lane = threadIdx.x & 31;
    const int wid  = threadIdx.x >> 5;

    for (int l = 0; l < L_SEQ; ++l) {
        const float v = ym[l];
        float sx = v, sy = v * v;
        #pragma unroll
        for (int off = 16; off > 0; off >>= 1) {
            sx += __shfl_down(sx, off);
            sy += __shfl_down(sy, off);
        }
        if (lane == 0) { redx[wid] = sx; redy[wid] = sy; }
        __syncthreads();
        if (wid == 0) {
            float tx = (lane < 16) ? redx[lane] : 0.0f;
            float ty = (lane < 16) ? redy[lane] : 0.0f;
            #pragma unroll
            for (int off = 8; off > 0; off >>= 1) {
                tx += __shfl_down(tx, off);
                ty += __shfl_down(ty, off);
            }
            if (lane == 0) { statx = tx; staty = ty; }
        }
        __syncthreads();
        const float mu  = statx * (1.0f / D_INNER);
        const float var = staty * (1.0f / D_INNER) - mu * mu;
        const float nrm = (v - mu) * rsqrtf(var + LN_EPS) * gw + gb;

        const size_t m = (size_t)(bb * L_SEQ + l) * 1024 + hw;
        const float z = zg[m * D_INNER + d];
        yg[m * D_INNER + d] = nrm * fast_silu(z);
        __syncthreads();
    }
}

// =====================================================================
// Launch
// =====================================================================
extern "C" void kernel_launch(void* const* d_in, const int* in_sizes, int n_in,
                              void* d_out, int out_size, void* d_ws, size_t ws_size,
                              hipStream_t stream) {
    (void)in_sizes; (void)n_in; (void)out_size; (void)ws_size;

    const float* x      = (const float*)d_in[0];
    const float* Win    = (const float*)d_in[1];
    const float* convw  = (const float*)d_in[2];
    const float* convb  = (const float*)d_in[3];
    const float* Wx     = (const float*)d_in[4];
    const float* dtw    = (const float*)d_in[5];
    const float* dtb    = (const float*)d_in[6];
    const float* Alogs  = (const float*)d_in[7];
    const float* Ds     = (const float*)d_in[8];
    const float* lnw    = (const float*)d_in[9];
    const float* lnb    = (const float*)d_in[10];
    const float* Wout   = (const float*)d_in[11];
    float* out = (float*)d_out;

    float* ws   = (float*)d_ws;
    const size_t SZ_MD = (size_t)M_ROWS * D_INNER;
    float* xc    = ws;                                      // [M,512]
    float* zg    = xc + SZ_MD;                              // [M,512]
    float* xcs   = zg + SZ_MD;                              // [P*L,512]
    float* xdbl0 = xcs + SZ_MD;                             // [M,48]
    float* xdbl1 = xdbl0 + (size_t)M_ROWS * C_DBL;          // [M,48]
    float* yg    = xc;                                      // reuse xc after conv

    k_inproj_gemm<<<dim3(M_ROWS / 64, (2 * D_INNER) / 128), 256, 0, stream>>>(x, Win, xc, zg);
    k_conv_dw<<<M_ROWS, D_INNER, 0, stream>>>(xc, convw, convb, xcs);
    k_xdbl_gemm<<<M_ROWS / 64, 256, 0, stream>>>(xcs, Wx, xdbl0, xdbl1);
    k_scan_ln_gate<<<P_ROWS, D_INNER, 0, stream>>>(
        xcs, xdbl0, xdbl1, dtw, dtb, Alogs, Ds, lnw, lnb, zg, yg);
    k_outproj_gemm<<<dim3(M_ROWS / 64, D_MODEL / 128), 256, 0, stream>>>(yg, Wout, out);
}